// SphericalHarmonicTransform_72327249264648
// MI455X (gfx1250) — compile-verified
//
#include <hip/hip_runtime.h>

#define L_MAX   8
#define RCUT    5.0f
#define NPAIR   45          // (l,m) pairs, l<=8, 0<=m<=l
#define TPB     256
#define NB_MAX  512
#define WS_STRIDE 96        // 90 partial sums padded to 6 x 16 columns

typedef __attribute__((ext_vector_type(2))) float v2f;
typedef __attribute__((ext_vector_type(8))) float v8f;

__device__ __forceinline__ double dfact(int n) {
    double r = 1.0;
    for (int i = 2; i <= n; ++i) r *= (double)i;
    return r;  // fully constant-folded after unrolling (all call sites constant)
}

// ---------------------------------------------------------------------------
// Kernel 1: per-point polynomial evaluation + hierarchical deterministic
// reduction to per-block partials ws[block][0..89] ({Re,Im} per (l,m) pair).
// Compute-bound fp32 VALU kernel; accumulators live entirely in VGPRs.
// ---------------------------------------------------------------------------
__global__ void __launch_bounds__(TPB)
sht_partials(const float* __restrict__ pos, float* __restrict__ ws, int npts)
{
    float accR[NPAIR], accI[NPAIR];
    #pragma unroll
    for (int j = 0; j < NPAIR; ++j) { accR[j] = 0.f; accI[j] = 0.f; }

    const int tid    = blockIdx.x * TPB + threadIdx.x;
    const int stride = gridDim.x * TPB;

    for (int idx = tid; idx < npts; idx += stride) {
        const float px = pos[3 * idx + 0];
        const float py = pos[3 * idx + 1];
        const float pz = pos[3 * idx + 2];

        const float n2   = px * px + py * py + pz * pz;
        const float nrm  = sqrtf(n2);
        const bool  valid = nrm > 0.f;
        const float safe = valid ? nrm : 1.f;
        float cut = 0.5f * (__cosf(nrm * (float)(3.14159265358979323846 / 5.0)) + 1.f);
        cut = (nrm > RCUT) ? 0.f : cut;
        cut = valid ? cut : 0.f;
        const float inv_safe = __frcp_rn(safe);

        const float xpr = -0.5f * px, xpi = -0.5f * py;   // xp
        const float xmr =  0.5f * px, xmi = -0.5f * py;   // xm
        const float x0  = pz;

        // Power tables replicating reference powcmplx semantics exactly:
        // n==2 is the exact square; n>=3 uses the sequential-update chain
        // (imaginary part uses the *already updated* real part).
        float z1r[9], z1i[9], z2r[9], z2i[9];
        z1r[0] = 1.f; z1i[0] = 0.f; z2r[0] = 1.f; z2i[0] = 0.f;
        z1r[1] = xpr; z1i[1] = xpi; z2r[1] = xmr; z2i[1] = xmi;
        {
            float cr = xpr, ci = xpi;
            #pragma unroll
            for (int n = 2; n <= L_MAX; ++n) {
                float nr = xpr * cr - xpi * ci;
                float ni = xpr * ci + xpi * nr;   // uses updated nr (reference bug)
                cr = nr; ci = ni;
                z1r[n] = cr; z1i[n] = ci;
            }
            z1r[2] = xpr * xpr - xpi * xpi;       // n==2 special case overrides
            z1i[2] = 2.f * xpr * xpi;
        }
        {
            float cr = xmr, ci = xmi;
            #pragma unroll
            for (int n = 2; n <= L_MAX; ++n) {
                float nr = xmr * cr - xmi * ci;
                float ni = xmr * ci + xmi * nr;
                cr = nr; ci = ni;
                z2r[n] = cr; z2i[n] = ci;
            }
            z2r[2] = xmr * xmr - xmi * xmi;
            z2i[2] = 2.f * xmr * xmi;
        }
        float s1r2[9], s2i2[9];
        #pragma unroll
        for (int k = 0; k <= L_MAX; ++k) {
            s1r2[k] = z1r[k] * z1r[k];
            s2i2[k] = z2i[k] * z2i[k];
        }

        float sinv_l = 1.f;   // inv_safe^l
        #pragma unroll
        for (int l = 0; l <= L_MAX; ++l) {
            const float w_l = cut * sinv_l *
                (float)__builtin_sqrt((2.0 * l + 1.0) / (4.0 * 3.14159));
            const float wx = w_l * x0;
            #pragma unroll
            for (int m = 0; m <= l; ++m) {
                float sr = 0.f, si = 0.f;
                #pragma unroll
                for (int p = 0; p <= l; ++p) {
                    const int q = p - m;
                    const int s = l - p - q;
                    if (q >= 0 && s >= 0) {
                        const float cf = (float)(1.0 / (dfact(p) * dfact(q) * dfact(s)));
                        const float zr = s1r2[p] - s2i2[q];                       // z1r^2 - z2i^2
                        const float zi = z1r[p] * z2i[q] + z1i[p] * z2r[q];
                        sr = fmaf(zr, cf, sr);
                        si = fmaf(zi, cf, si);
                    }
                }
                const float f  = (float)__builtin_sqrt(dfact(l + m) * dfact(l - m));
                const float ww = f * wx;
                const int   j  = l * (l + 1) / 2 + m;
                accR[j] = fmaf(sr, ww, accR[j]);
                accI[j] = fmaf(si, ww, accI[j]);
            }
            sinv_l *= inv_safe;
        }
    }

    // wave32 shuffle reduction
    #pragma unroll
    for (int j = 0; j < NPAIR; ++j) {
        float r = accR[j], im = accI[j];
        #pragma unroll
        for (int off = 16; off > 0; off >>= 1) {
            r  += __shfl_down(r,  off, 32);
            im += __shfl_down(im, off, 32);
        }
        accR[j] = r; accI[j] = im;
    }

    __shared__ float red[TPB / 32][2 * NPAIR];
    const int lane = threadIdx.x & 31;
    const int wave = threadIdx.x >> 5;
    if (lane == 0) {
        #pragma unroll
        for (int j = 0; j < NPAIR; ++j) {
            red[wave][2 * j]     = accR[j];
            red[wave][2 * j + 1] = accI[j];
        }
    }
    __syncthreads();
    if (threadIdx.x < 2 * NPAIR) {
        float s = 0.f;
        #pragma unroll
        for (int w = 0; w < TPB / 32; ++w) s += red[w][threadIdx.x];
        ws[blockIdx.x * WS_STRIDE + threadIdx.x] = s;
    }
}

// ---------------------------------------------------------------------------
// Kernel 2: cross-block reduction as a matmul-with-ones on the fp32 WMMA pipe:
//   D(16x16) = A(16x4, all ones) x B(4 partial rows x 16 cols) + C
// keeps the reduction in full fp32 (V_WMMA_F32_16X16X4_F32), then one lane
// applies the sqrt(2)*(-1)^m mirror mapping into the 81 outputs.
// One wave of 32 threads => EXEC all ones as WMMA requires.
// ---------------------------------------------------------------------------
__global__ void __launch_bounds__(32)
sht_finalize(const float* __restrict__ ws, float* __restrict__ out, int nb)
{
    __shared__ float totals[WS_STRIDE];
    const int lane = threadIdx.x;          // 0..31
    const int col  = lane & 15;            // column within 16-wide chunk
    const int rsel = (lane >> 4) & 1;      // lanes 0-15 carry rows {0,2}, 16-31 rows {1,3}

    for (int c = 0; c < WS_STRIDE / 16; ++c) {
        v8f acc = {0.f, 0.f, 0.f, 0.f, 0.f, 0.f, 0.f, 0.f};
        v2f a;  a.x = 1.f; a.y = 1.f;      // A = all-ones 16x4
        for (int b = 0; b < nb; b += 4) {
            v2f bm;
            bm.x = ws[(b + rsel)     * WS_STRIDE + c * 16 + col];
            bm.y = ws[(b + rsel + 2) * WS_STRIDE + c * 16 + col];
            acc = __builtin_amdgcn_wmma_f32_16x16x4_f32(
                /*neg_a=*/false, a, /*neg_b=*/false, bm,
                /*c_mod=*/(short)0, acc, /*reuse_a=*/false, /*reuse_b=*/false);
        }
        // D[m][n] = sum_k B[k][n] for every m; row M=0 lives in VGPR0, lanes 0..15
        if (lane < 16) totals[c * 16 + lane] = acc[0];
    }
    __syncthreads();

    if (lane == 0) {
        for (int l = 0; l <= L_MAX; ++l) {
            const int base = l * l + l;
            for (int m = 0; m <= l; ++m) {
                const int   j  = l * (l + 1) / 2 + m;
                const float cr = totals[2 * j];
                const float ci = totals[2 * j + 1];
                if (m == 0) {
                    out[base] = cr;
                } else {
                    const float fac = (m & 1) ? -1.41421356237309515f
                                              :  1.41421356237309515f;
                    out[base - m] = fac * ci;
                    out[base + m] = fac * cr;
                }
            }
        }
    }
}

extern "C" void kernel_launch(void* const* d_in, const int* in_sizes, int n_in,
                              void* d_out, int out_size, void* d_ws, size_t ws_size,
                              hipStream_t stream)
{
    const float* pos = (const float*)d_in[0];
    const int npts = in_sizes[0] / 3;
    float* out = (float*)d_out;
    float* ws  = (float*)d_ws;

    // number of partial blocks, limited by workspace, multiple of 4 for the
    // 4-row WMMA reduction chunks
    int nb = (int)(ws_size / (WS_STRIDE * sizeof(float)));
    if (nb > NB_MAX) nb = NB_MAX;
    nb &= ~3;
    if (nb < 4) nb = 4;

    sht_partials<<<nb, TPB, 0, stream>>>(pos, ws, npts);
    sht_finalize<<<1, 32, 0, stream>>>(ws, out, nb);
}